// SupConLoss_31172872634655
// MI455X (gfx1250) — compile-verified
//
#include <hip/hip_runtime.h>

#define B_    2048
#define D_    1024
#define N_    4096
#define NCLS  100
#define CENT  10
#define SPLIT 8            // column splits of the logits GEMM (parallelism)

typedef __attribute__((ext_vector_type(2))) float v2f;
typedef __attribute__((ext_vector_type(8))) float v8f;
typedef __attribute__((ext_vector_type(4))) int   v4i;

// workspace layout (floats)
#define WS_ACC_LOSS 0
#define WS_ACC_AVG  1
#define WS_CNT      16
#define WS_CLS      128                    // classsum [100][1024]
#define WS_TOT      (128 + NCLS * D_)      // totalavg [1024]
#define WS_PART     (128 + NCLS * D_ + D_) // partials: 4 arrays [SPLIT][N_]
#define WS_FLOATS   (WS_PART + 4 * SPLIT * N_)

#if defined(__has_builtin)
#if __has_builtin(__builtin_amdgcn_global_load_async_to_lds_b128) && \
    __has_builtin(__builtin_amdgcn_s_wait_asynccnt)
#define USE_ASYNC 1
#endif
#endif
#ifndef USE_ASYNC
#define USE_ASYNC 0
#endif

__device__ __forceinline__ const float* anchor_row(const float* f, int i) {
  // anchor = cat(features[:,0], features[:,1]); features is [B,2,D]
  int idx = (i < B_) ? (i * 2) : ((i - B_) * 2 + 1);
  return f + (size_t)idx * D_;
}

__device__ __forceinline__ int label2(const int* labels, int i) {
  return labels[(i < B_) ? i : (i - B_)];
}

__global__ void zero_ws(float* ws, int n) {
  int i = blockIdx.x * blockDim.x + threadIdx.x;
  if (i < n) ws[i] = 0.f;
}

__global__ void class_count(const int* labels, float* cnt) {
  int i = blockIdx.x * blockDim.x + threadIdx.x;
  if (i < B_) atomicAdd(&cnt[labels[i]], 1.0f);
}

__global__ void class_sums(const float* avg, float* cls, float* tot) {
  int i = blockIdx.x * blockDim.x + threadIdx.x;
  if (i < NCLS * D_) {
    int c = i / D_, d = i % D_;
    float s = 0.f;
    for (int t = 0; t < CENT; ++t) s += avg[(size_t)(c * CENT + t) * D_ + d];
    cls[i] = s;
  } else if (i < NCLS * D_ + D_) {
    int d = i - NCLS * D_;
    float s = 0.f;
    for (int j = 0; j < NCLS * CENT; ++j) s += avg[(size_t)j * D_ + d];
    tot[d] = s;
  }
}

// average_loss collapsed via linearity: sum_i anchor_i . (2*classsum[lbl_i]-totalavg)
__global__ void avg_branch(const float* feat, const int* labels,
                           const float* cls, const float* tot, float* acc) {
  __shared__ float red[256];
  int i = blockIdx.x;
  int lbl = labels[(i < B_) ? i : (i - B_)];
  const float* a  = anchor_row(feat, i);
  const float* cs = cls + (size_t)lbl * D_;
  float p = 0.f;
  for (int d = threadIdx.x; d < D_; d += 256)
    p += a[d] * (2.f * cs[d] - tot[d]);
  red[threadIdx.x] = p;
  __syncthreads();
  for (int s = 128; s > 0; s >>= 1) {
    if (threadIdx.x < (unsigned)s) red[threadIdx.x] += red[threadIdx.x + s];
    __syncthreads();
  }
  if (threadIdx.x == 0) atomicAdd(acc, red[0]);
}

#if USE_ASYNC
typedef __attribute__((address_space(1))) v4i* glb_vp;  // global int4*
typedef __attribute__((address_space(3))) v4i* lds_vp;  // LDS int4*

// Issue 6 async b128 global->LDS transfers per thread (A: 4, B: 2).
__device__ __forceinline__ void stage_async(const float* feat, float* Ab, float* Bb,
                                            int row0, int jc, int kb, int tid) {
#pragma unroll
  for (int p = 0; p < 4; ++p) {
    const int q = tid + 256 * p;
    const int r = q >> 3, kk = (q & 7) << 2;
    const float* g = anchor_row(feat, row0 + r) + kb + kk;
    __builtin_amdgcn_global_load_async_to_lds_b128(
        (glb_vp)(v4i*)(float*)g, (lds_vp)(v4i*)&Ab[r * 32 + kk], 0, 0);
  }
#pragma unroll
  for (int p = 0; p < 2; ++p) {
    const int q = tid + 256 * p;
    const int r = q >> 3, kk = (q & 7) << 2;
    const float* g = anchor_row(feat, jc + r) + kb + kk;
    __builtin_amdgcn_global_load_async_to_lds_b128(
        (glb_vp)(v4i*)(float*)g, (lds_vp)(v4i*)&Bb[r * 32 + kk], 0, 0);
  }
}
#else
__device__ __forceinline__ void stage_sync(const float* feat, float* Ab, float* Bb,
                                           int row0, int jc, int kb, int tid) {
#pragma unroll
  for (int p = 0; p < 4; ++p) {
    const int q = tid + 256 * p;
    const int r = q >> 3, kk = (q & 7) << 2;
    *(float4*)&Ab[r * 32 + kk] = *(const float4*)(anchor_row(feat, row0 + r) + kb + kk);
  }
#pragma unroll
  for (int p = 0; p < 2; ++p) {
    const int q = tid + 256 * p;
    const int r = q >> 3, kk = (q & 7) << 2;
    *(float4*)&Bb[r * 32 + kk] = *(const float4*)(anchor_row(feat, jc + r) + kb + kk);
  }
}
#endif

// 32-deep K stage: 8 substeps x 4 column fragments of V_WMMA_F32_16X16X4_F32
__device__ __forceinline__ void compute_stage(v8f c[4], const float* Ab, const float* Bb,
                                              int wrow, int l15, int half) {
#pragma unroll
  for (int s4 = 0; s4 < 8; ++s4) {
    const int k0 = s4 * 4 + half * 2;  // lanes 0-15: K={k,k+1}; lanes 16-31: K={k+2,k+3}
    v2f a;
    a.x = Ab[(wrow + l15) * 32 + k0];
    a.y = Ab[(wrow + l15) * 32 + k0 + 1];
#pragma unroll
    for (int f = 0; f < 4; ++f) {
      v2f b;
      b.x = Bb[(f * 16 + l15) * 32 + k0];
      b.y = Bb[(f * 16 + l15) * 32 + k0 + 1];
      c[f] = __builtin_amdgcn_wmma_f32_16x16x4_f32(
               false, a, false, b, (short)0, c[f], false, false);
    }
  }
}

// Fused logits GEMM (f32 WMMA) + flash-style online per-row reductions.
// grid = (N_/128 row strips) x (SPLIT column groups of 512 cols).
// 8 waves per block; each wave owns 16 rows (wave-private softmax state).
__launch_bounds__(256)
__global__ void supcon_main(const float* feat, const int* labels,
                            const float* cnt, float* part) {
  __shared__ float sA[2][128 * 32];  // 32 KB (double-buffered row tile)
  __shared__ float sB[2][64 * 32];   // 16 KB (double-buffered col tile)
  __shared__ float sCnt[128];
  __shared__ int   sRowLbl[128];
  __shared__ int   sColLbl[64];
  __shared__ float sM[128], sS[128], sSml[128], sPos[128];

  const int tid  = threadIdx.x;
  const int wave = tid >> 5;
  const int lane = tid & 31;
  const int half = lane >> 4;
  const int l15  = lane & 15;
  const int row0 = blockIdx.x * 128;
  const int col0 = blockIdx.y * (N_ / SPLIT);
  const int wrow = wave * 16;

  if (tid < 128) {
    sCnt[tid]    = (tid < NCLS) ? cnt[tid] : 1.0f;
    sRowLbl[tid] = label2(labels, row0 + tid);
    sM[tid] = -3.0e38f; sS[tid] = 0.f; sSml[tid] = 0.f; sPos[tid] = 0.f;
  }
  __syncthreads();

  const float invT = 10.0f;  // 1 / TEMPERATURE

  for (int jt = 0; jt < (N_ / SPLIT) / 64; ++jt) {
    const int jc = col0 + jt * 64;
    __syncthreads();  // prev epilogue done; safe to refill sColLbl / buffers
    if (tid < 64) sColLbl[tid] = label2(labels, jc + tid);

    v8f c[4] = {};

#if USE_ASYNC
    stage_async(feat, &sA[0][0], &sB[0][0], row0, jc, 0, tid);
    int pb = 0;
    for (int kb = 0; kb < D_; kb += 32, pb ^= 1) {
      if (kb + 32 < D_) {
        stage_async(feat, &sA[pb ^ 1][0], &sB[pb ^ 1][0], row0, jc, kb + 32, tid);
        __builtin_amdgcn_s_wait_asynccnt(6);  // in-order: current 6 done
      } else {
        __builtin_amdgcn_s_wait_asynccnt(0);
      }
      __syncthreads();
      compute_stage(c, &sA[pb][0], &sB[pb][0], wrow, l15, half);
      __syncthreads();
    }
#else
    for (int kb = 0; kb < D_; kb += 32) {
      __syncthreads();
      stage_sync(feat, &sA[0][0], &sB[0][0], row0, jc, kb, tid);
      __syncthreads();
      compute_stage(c, &sA[0][0], &sB[0][0], wrow, l15, half);
    }
    __syncthreads();
#endif

    // online per-row update over this 64-column tile
#pragma unroll 1
    for (int r = 0; r < 8; ++r) {
      const int li   = wrow + r + half * 8;  // C layout: M = r (+8 for hi half)
      const int grow = row0 + li;
      const int lbli = sRowLbl[li];
      float lv[4];
#pragma unroll
      for (int f = 0; f < 4; ++f) lv[f] = c[f][r] * invT;
      float tmax = fmaxf(fmaxf(lv[0], lv[1]), fmaxf(lv[2], lv[3]));
#pragma unroll
      for (int m = 1; m < 16; m <<= 1)
        tmax = fmaxf(tmax, __shfl_xor(tmax, m, 32));
      const float mold = sM[li];
      const float newm = fmaxf(mold, tmax);
      float sl = 0.f, sml = 0.f, pl = 0.f;
#pragma unroll
      for (int f = 0; f < 4; ++f) {
        const int gj   = jc + f * 16 + l15;
        const int lblj = sColLbl[f * 16 + l15];
        const float lm = (gj == grow) ? 0.f : 1.f;   // logits_mask (no self)
        const float mk = (lblj == lbli) ? lm : 0.f;  // positive mask
        float per = sCnt[lblj] - mk;
        per = (per < 1e-6f) ? 1.f : per;
        sl  += lm * __expf(lv[f] - newm) / per;
        sml += mk * lv[f];
        pl  += mk;
      }
#pragma unroll
      for (int m = 1; m < 16; m <<= 1) {
        sl  += __shfl_xor(sl,  m, 32);
        sml += __shfl_xor(sml, m, 32);
        pl  += __shfl_xor(pl,  m, 32);
      }
      if (l15 == 0) {
        sS[li]   = sS[li] * __expf(mold - newm) + sl;
        sM[li]   = newm;
        sSml[li] += sml;
        sPos[li] += pl;
      }
    }
  }

  __syncthreads();
  if (tid < 128) {
    const int idx = blockIdx.y * N_ + row0 + tid;
    part[idx]                  = sM[tid];
    part[SPLIT * N_ + idx]     = sS[tid];
    part[2 * SPLIT * N_ + idx] = sSml[tid];
    part[3 * SPLIT * N_ + idx] = sPos[tid];
  }
}

// Combine per-split flash partials and accumulate the per-row loss term.
__global__ void merge_loss(const float* part, float* acc) {
  int row = blockIdx.x * blockDim.x + threadIdx.x;
  if (row >= N_) return;
  const float* PM = part;
  const float* PS = part + SPLIT * N_;
  const float* PL = part + 2 * SPLIT * N_;
  const float* PP = part + 3 * SPLIT * N_;
  float m = -3.0e38f;
  for (int s = 0; s < SPLIT; ++s) m = fmaxf(m, PM[s * N_ + row]);
  float ss = 0.f, sml = 0.f, pos = 0.f;
  for (int s = 0; s < SPLIT; ++s) {
    ss  += PS[s * N_ + row] * __expf(PM[s * N_ + row] - m);
    sml += PL[s * N_ + row];
    pos += PP[s * N_ + row];
  }
  const float posc = (pos < 1e-6f) ? 1.f : pos;
  // sum(mask*log_prob) = sml - pos*max - pos*log(denom)
  const float t = (sml - pos * m - pos * __logf(ss)) / posc;
  atomicAdd(acc, t);
}

__global__ void finalize(const float* ws, float* out) {
  if (threadIdx.x == 0 && blockIdx.x == 0) {
    out[0] = -ws[WS_ACC_LOSS] / (float)N_;
    out[1] = -ws[WS_ACC_AVG] / ((float)N_ * (float)(NCLS * CENT));
  }
}

extern "C" void kernel_launch(void* const* d_in, const int* in_sizes, int n_in,
                              void* d_out, int out_size, void* d_ws, size_t ws_size,
                              hipStream_t stream) {
  const float* feat   = (const float*)d_in[0];
  const int*   labels = (const int*)d_in[1];
  const float* avg    = (const float*)d_in[2];
  float* ws  = (float*)d_ws;
  float* out = (float*)d_out;

  zero_ws<<<(WS_PART + 255) / 256, 256, 0, stream>>>(ws, WS_PART);
  class_count<<<(B_ + 255) / 256, 256, 0, stream>>>(labels, ws + WS_CNT);
  class_sums<<<(NCLS * D_ + D_ + 255) / 256, 256, 0, stream>>>(
      avg, ws + WS_CLS, ws + WS_TOT);
  avg_branch<<<N_, 256, 0, stream>>>(feat, labels, ws + WS_CLS, ws + WS_TOT,
                                     ws + WS_ACC_AVG);
  supcon_main<<<dim3(N_ / 128, SPLIT), 256, 0, stream>>>(feat, labels,
                                                         ws + WS_CNT, ws + WS_PART);
  merge_loss<<<(N_ + 255) / 256, 256, 0, stream>>>(ws + WS_PART, ws + WS_ACC_LOSS);
  finalize<<<1, 1, 0, stream>>>(ws, out);
}